// KgembKModel_9955734192251
// MI455X (gfx1250) — compile-verified
//
#include <hip/hip_runtime.h>

// ---------------------------------------------------------------------------
// Problem sizes (bert-base KG-embedding fusion)
// ---------------------------------------------------------------------------
static constexpr int Bsz = 32, Ssz = 512, Esz = 8, Hsz = 768;
static constexpr int KG  = 100, MH = 1000;
static constexpr int BE  = Bsz * Esz;          // 256 entity rows total

typedef float v2f __attribute__((ext_vector_type(2)));
typedef float v8f __attribute__((ext_vector_type(8)));

// ---------------------------------------------------------------------------
// f32 WMMA GEMM:  C[256,NDIM] = act( A[256,KDIM] @ W[KDIM,NDIM] + bias )
// One wave32 computes one 16x16 tile of C with V_WMMA_F32_16X16X4_F32,
// stepping K by 4 each iteration.
//
// ISA lane layouts (CDNA5 ISA):
//   A 16x4 : lanes 0-15 hold M=lane, {VGPR0,VGPR1} = K={0,1};
//            lanes 16-31 hold M=lane-16, K={2,3}.
//   B 4x16 : lanes 0-15 hold N=lane, K={0,1}; lanes 16-31 K={2,3}.
//   C/D    : VGPR r -> M = r (lanes 0-15) / r+8 (lanes 16-31), N = lane&15.
// EXEC must be all-ones for WMMA: tile bounds use clamped loads and
// predicated stores only (no divergent branch spans the WMMA).
// ---------------------------------------------------------------------------
template <int KDIM, int NDIM, bool RELU>
__global__ __launch_bounds__(256) void gemm256_wmma(
    const float* __restrict__ A,     // [256, KDIM] row-major
    const float* __restrict__ W,     // [KDIM, NDIM] row-major
    const float* __restrict__ bias,  // [NDIM]
    float* __restrict__ C)           // [256, NDIM] row-major
{
    constexpr int NT    = (NDIM + 15) / 16;
    constexpr int TILES = (BE / 16) * NT;

    const int wave = threadIdx.x >> 5;
    const int tile = blockIdx.x * (blockDim.x >> 5) + wave;
    if (tile >= TILES) return;               // wave-uniform: EXEC stays full

    const int mT = tile / NT;
    const int nT = tile % NT;

    const int lane = threadIdx.x & 31;
    const int hi   = (lane >> 4) & 1;        // 0: lanes 0-15, 1: lanes 16-31
    const int l    = lane & 15;

    const int m  = mT * 16 + l;              // A row this lane feeds
    const int n  = nT * 16 + l;              // B/D column this lane feeds
    const int nc = (n < NDIM) ? n : (NDIM - 1);  // clamp keeps loads in-range

    const float* __restrict__ Arow = A + (long long)m * KDIM;

    v8f acc = {};
    #pragma unroll 2
    for (int k = 0; k < KDIM; k += 4) {
        const int ka = k + (hi ? 2 : 0);
        v2f a, b;
        a.x = Arow[ka];
        a.y = Arow[ka + 1];
        b.x = W[(long long)ka * NDIM + nc];
        b.y = W[(long long)(ka + 1) * NDIM + nc];
        // 8 args: (neg_a, A, neg_b, B, c_mod, C, reuse_a, reuse_b)
        acc = __builtin_amdgcn_wmma_f32_16x16x4_f32(
            false, a, false, b, (short)0, acc, false, false);
    }

    const float bc = bias[nc];
    #pragma unroll
    for (int r = 0; r < 8; ++r) {
        const int row = mT * 16 + r + (hi ? 8 : 0);
        float v = acc[r] + bc;
        if (RELU) v = fmaxf(v, 0.0f);
        if (n < NDIM) C[(long long)row * NDIM + n] = v;   // predicated store
    }
}

// ---------------------------------------------------------------------------
// Bandwidth kernel: out[b,s,:] = word_embedding[ids[b,s]] +
//                                sum_e mask[b,e,s] * ent[b*E+e,:]
// One 192-thread block per (b,s) row; one float4 per thread (b128 traffic).
// Mask is ~2% dense, so the ent-row loads are usually skipped.
// ---------------------------------------------------------------------------
__global__ __launch_bounds__(192) void gather_scatter(
    const int*   __restrict__ ids,   // [B,S]
    const float* __restrict__ mask,  // [B,E,S]
    const float* __restrict__ wemb,  // [V,H]
    const float* __restrict__ ent,   // [B*E,H]
    float*       __restrict__ out)   // [B,S,H]
{
    const int row = blockIdx.x;           // b*S + s
    const int b   = row >> 9;             // / 512
    const int s   = row & (Ssz - 1);
    const int t   = threadIdx.x;          // 0..191 -> one float4 of H=768

    const long long wid = ids[row];
    float4 acc = reinterpret_cast<const float4*>(wemb + wid * Hsz)[t];

    #pragma unroll
    for (int e = 0; e < Esz; ++e) {
        const int   er   = b * Esz + e;
        const float mval = mask[(long long)er * Ssz + s];
        if (mval != 0.0f) {
            float4 v = reinterpret_cast<const float4*>(ent + (long long)er * Hsz)[t];
            acc.x = fmaf(mval, v.x, acc.x);
            acc.y = fmaf(mval, v.y, acc.y);
            acc.z = fmaf(mval, v.z, acc.z);
            acc.w = fmaf(mval, v.w, acc.w);
        }
    }
    reinterpret_cast<float4*>(out + (long long)row * Hsz)[t] = acc;
}

// ---------------------------------------------------------------------------
// Launch: GEMM1 (relu) -> GEMM2 -> gather/scatter-add, all on `stream`.
// Workspace: Hact [256,1000] then ent [256,768] (~1.8 MB).
// ---------------------------------------------------------------------------
extern "C" void kernel_launch(void* const* d_in, const int* in_sizes, int n_in,
                              void* d_out, int out_size, void* d_ws, size_t ws_size,
                              hipStream_t stream) {
    const int*   ids  = (const int*)  d_in[0];
    const float* eemb = (const float*)d_in[1];   // [B,E,100] == [256,100]
    const float* mask = (const float*)d_in[2];   // [B,E,S]
    const float* wemb = (const float*)d_in[3];   // [V,H]
    const float* W1   = (const float*)d_in[4];   // [100,1000]
    const float* b1   = (const float*)d_in[5];   // [1000]
    const float* W2   = (const float*)d_in[6];   // [1000,768]
    const float* b2   = (const float*)d_in[7];   // [768]

    float* Hact = (float*)d_ws;                  // [256,1000]
    float* ent  = Hact + (size_t)BE * MH;        // [256,768]

    {
        constexpr int TILES = (BE / 16) * ((MH + 15) / 16);   // 16*63 = 1008
        gemm256_wmma<KG, MH, true>
            <<<(TILES + 7) / 8, 256, 0, stream>>>(eemb, W1, b1, Hact);
    }
    {
        constexpr int TILES = (BE / 16) * ((Hsz + 15) / 16);  // 16*48 = 768
        gemm256_wmma<MH, Hsz, false>
            <<<(TILES + 7) / 8, 256, 0, stream>>>(Hact, W2, b2, ent);
    }
    gather_scatter<<<Bsz * Ssz, 192, 0, stream>>>(
        ids, mask, wemb, ent, (float*)d_out);
}